// GeneReconModel_57982058496458
// MI455X (gfx1250) — compile-verified
//
#include <hip/hip_runtime.h>

// ---- problem constants (match reference) ----
#define N_LEAVES 65536
#define NWAVES   24
#define WC       2048
#define KSPL     3
#define SSPEC    128
#define CTOT     (N_LEAVES + NWAVES * WC)   // 114688
#define GFAM     2048
#define E_ITERS  50
#define EPSV     1e-30f

typedef __attribute__((ext_vector_type(2))) float v2f;
typedef __attribute__((ext_vector_type(8))) float v8f;

// ---------------- kernel 0: rates + extinction fixed point ----------------
__global__ void k_rates(const float* __restrict__ theta,
                        float* __restrict__ denom,
                        float* __restrict__ params) {
    __shared__ float sh[SSPEC];
    int s = threadIdx.x;
    float D = exp2f(theta[0]);
    float L = exp2f(theta[1]);
    float T = exp2f(theta[2]);
    float norm = D + L + T + 1.0f;
    float e = L / norm;
    for (int it = 0; it < E_ITERS; ++it) {
        sh[s] = e;
        __syncthreads();
        float sum = 0.0f;
        for (int i = 0; i < SSPEC; ++i) sum += sh[i];   // deterministic serial order
        float mean = sum / (float)SSPEC;
        __syncthreads();
        e = (L + D * e * e + T * mean * e) / norm;
    }
    sh[s] = e;
    __syncthreads();
    float sum = 0.0f;
    for (int i = 0; i < SSPEC; ++i) sum += sh[i];
    float emean = sum / (float)SSPEC;
    denom[s] = 1.0f + D * e + T * emean;
    if (s == 0) {
        // term = cA*Pl*Pr + cB*(Pl*sumPr + Pr*sumPl)  (exact rearrangement of pib form)
        params[0] = (1.0f + D) - 2.0f * T / (float)(SSPEC - 1);  // cA
        params[1] = T / (float)(SSPEC - 1);                       // cB
    }
}

// ---------------- kernel 1: zero P and ls (contiguous) ----------------
__global__ void k_zero(float* __restrict__ buf, long long n) {
    long long i = (long long)blockIdx.x * blockDim.x + threadIdx.x;
    long long stride = (long long)gridDim.x * blockDim.x;
    for (; i < n; i += stride) buf[i] = 0.0f;
}

// ---------------- kernel 2: leaf one-hot scatter ----------------
__global__ void k_leaves(float* __restrict__ P, const int* __restrict__ leaf_species) {
    int i = blockIdx.x * blockDim.x + threadIdx.x;
    int sp = leaf_species[i];
    P[(long long)i * SSPEC + sp] = 1.0f;
}

// ---------------- kernel 3: one topological wave. one wave32 per clade ----------------
__global__ __launch_bounds__(256) void k_wave(
    float* __restrict__ P, float* __restrict__ ls,
    const float* __restrict__ denom, const float* __restrict__ params,
    const float* __restrict__ logw, const int* __restrict__ left,
    const int* __restrict__ right, int w)
{
    int tid   = blockIdx.x * blockDim.x + threadIdx.x;
    int clade = tid >> 5;          // exact grid: 2048 clades * 32 lanes
    int lane  = tid & 31;

    long long base = ((long long)w * WC + clade) * KSPL;
    int   li[KSPL], ri[KSPL];
    float lw[KSPL];
    for (int k = 0; k < KSPL; ++k) {
        li[k] = left[base + k];
        ri[k] = right[base + k];
        lw[k] = logw[base + k];
    }
    // softmax over the K=3 split log-weights
    float mx = fmaxf(lw[0], fmaxf(lw[1], lw[2]));
    float e0 = expf(lw[0] - mx), e1 = expf(lw[1] - mx), e2 = expf(lw[2] - mx);
    float inv = 1.0f / (e0 + e1 + e2);
    float wk[KSPL] = { e0 * inv, e1 * inv, e2 * inv };

    float cA = params[0];
    float cB = params[1];

    // rows belonging to the CURRENT wave must be seen as pre-update (== 0)
    int wlo = N_LEAVES + w * WC;
    int whi = wlo + WC;

    float a0 = 0.f, a1 = 0.f, a2 = 0.f, a3 = 0.f;
    for (int k = 0; k < KSPL; ++k) {
        float4 pl = make_float4(0.f, 0.f, 0.f, 0.f);
        float4 pr = make_float4(0.f, 0.f, 0.f, 0.f);
        if (!(li[k] >= wlo && li[k] < whi))   // wave-uniform branch
            pl = ((const float4*)(P + (long long)li[k] * SSPEC))[lane];
        if (!(ri[k] >= wlo && ri[k] < whi))
            pr = ((const float4*)(P + (long long)ri[k] * SSPEC))[lane];

        float sl = pl.x + pl.y + pl.z + pl.w;
        float sr = pr.x + pr.y + pr.z + pr.w;
        for (int off = 16; off >= 1; off >>= 1) {   // wave32 allreduce (ds path)
            sl += __shfl_xor(sl, off, 32);
            sr += __shfl_xor(sr, off, 32);
        }
        float wkk = wk[k];
        a0 += wkk * (cA * pl.x * pr.x + cB * (pl.x * sr + pr.x * sl));
        a1 += wkk * (cA * pl.y * pr.y + cB * (pl.y * sr + pr.y * sl));
        a2 += wkk * (cA * pl.z * pr.z + cB * (pl.z * sr + pr.z * sl));
        a3 += wkk * (cA * pl.w * pr.w + cB * (pl.w * sr + pr.w * sl));
    }

    float4 d4 = ((const float4*)denom)[lane];
    a0 /= d4.x; a1 /= d4.y; a2 /= d4.z; a3 /= d4.w;

    float m = fmaxf(fmaxf(a0, a1), fmaxf(a2, a3));
    for (int off = 16; off >= 1; off >>= 1)
        m = fmaxf(m, __shfl_xor(m, off, 32));
    float safe = fmaxf(m, EPSV);

    long long id = (long long)N_LEAVES + (long long)w * WC + clade;
    float4 outv = make_float4(a0 / safe, a1 / safe, a2 / safe, a3 / safe);
    ((float4*)(P + id * SSPEC))[lane] = outv;
    if (lane == 0) ls[id] = logf(safe);
}

// ---------------- kernel 4: root row-sums via V_WMMA_F32_16X16X4_F32 ----------------
// 16 families per wave. A-tile = 16 gathered rows x 4 columns per step (32 steps = 128 cols).
// B = all-ones 4x16, so D[m][n] = rowsum(m) in EVERY column n (layout-robust readout).
__global__ __launch_bounds__(256) void k_root(
    const float* __restrict__ P, const float* __restrict__ ls,
    const int* __restrict__ root_ids, float* __restrict__ famll)
{
    int tid  = blockIdx.x * blockDim.x + threadIdx.x;
    int wave = tid >> 5;
    int lane = tid & 31;
    int m    = lane & 15;      // A-matrix row index held by this lane
    int half = lane >> 4;      // 0 -> K=0,1 (cols 4t,4t+1); 1 -> K=2,3 (cols 4t+2,4t+3)
    int famBase = wave * 16;

    int r = root_ids[famBase + m];
    const float* row = P + (long long)r * SSPEC;

    v2f bones; bones[0] = 1.0f; bones[1] = 1.0f;   // all-ones B
    v8f acc = {};
    for (int t = 0; t < 32; ++t) {
        float2 av = *((const float2*)(row + 4 * t + 2 * half));
        v2f a; a[0] = av.x; a[1] = av.y;
        acc = __builtin_amdgcn_wmma_f32_16x16x4_f32(
            /*neg_a=*/false, a, /*neg_b=*/false, bones,
            /*c_mod=*/(short)0, acc, /*reuse_a=*/false, /*reuse_b=*/false);
    }
    // D layout: lanes 0-15 VGPR j = M=j ; lanes 16-31 VGPR j = M=8+j.
    // With all-ones B every lane in a half holds identical data; lanes 0 and 16 write.
    if (m == 0) {
        for (int j = 0; j < 8; ++j) {
            int f  = famBase + half * 8 + j;
            int rr = root_ids[f];
            famll[f] = logf(acc[j] + EPSV) + ls[rr];
        }
    }
}

// ---------------- kernel 5: deterministic final reduction ----------------
__global__ void k_reduce(const float* __restrict__ famll, float* __restrict__ out) {
    __shared__ float sh[256];
    int t = threadIdx.x;
    float s = 0.0f;
    for (int i = t; i < GFAM; i += 256) s += famll[i];   // fixed order
    sh[t] = s;
    __syncthreads();
    for (int off = 128; off >= 1; off >>= 1) {
        if (t < off) sh[t] += sh[t + off];
        __syncthreads();
    }
    if (t == 0) out[0] = -sh[0];
}

extern "C" void kernel_launch(void* const* d_in, const int* in_sizes, int n_in,
                              void* d_out, int out_size, void* d_ws, size_t ws_size,
                              hipStream_t stream) {
    (void)in_sizes; (void)n_in; (void)out_size; (void)ws_size;

    const float* theta      = (const float*)d_in[0];
    const float* split_logw = (const float*)d_in[1];
    const int*   wave_left  = (const int*)d_in[2];
    const int*   wave_right = (const int*)d_in[3];
    const int*   leaf_spec  = (const int*)d_in[4];
    const int*   root_ids   = (const int*)d_in[5];

    // workspace layout (floats): P[C*S] | ls[C] | denom[S] | params[8] | famll[G]
    float* W      = (float*)d_ws;
    float* P      = W;
    float* ls     = P + (long long)CTOT * SSPEC;
    float* denom  = ls + CTOT;
    float* params = denom + SSPEC;
    float* famll  = params + 8;

    k_rates<<<1, SSPEC, 0, stream>>>(theta, denom, params);

    long long nz = (long long)CTOT * SSPEC + CTOT;   // P and ls are contiguous
    k_zero<<<2048, 256, 0, stream>>>(P, nz);
    k_leaves<<<N_LEAVES / 256, 256, 0, stream>>>(P, leaf_spec);

    for (int w = 0; w < NWAVES; ++w) {
        k_wave<<<(WC * 32) / 256, 256, 0, stream>>>(
            P, ls, denom, params, split_logw, wave_left, wave_right, w);
    }

    k_root<<<((GFAM / 16) * 32) / 256, 256, 0, stream>>>(P, ls, root_ids, famll);
    k_reduce<<<1, 256, 0, stream>>>(famll, (float*)d_out);
}